// Attention_18648747999393
// MI455X (gfx1250) — compile-verified
//
#include <hip/hip_runtime.h>
#include <stdint.h>

// WKV log-space scan for MI455X (gfx1250).
// B*D = 8192 independent recurrence chains; one channel per lane, one wave32
// per workgroup (no intra-block sync needed). k/v streamed into LDS with the
// Tensor Data Mover (tensor_load_to_lds), double-buffered, synchronized with
// s_wait_tensorcnt. Transcendentals via hardware v_exp_f32/v_log_f32.
// All outputs are written with non-temporal stores (streaming, never re-read).

#define WKV_EPS 1e-4f
constexpr int Bn = 8;
constexpr int Tn = 2048;
constexpr int Dn = 1024;
constexpr int BLK    = 32;              // channels per block == lanes per wave
constexpr int T_TILE = 64;              // timesteps per TDM tile
constexpr int NTILES = Tn / T_TILE;     // 32

typedef unsigned int u32x4 __attribute__((ext_vector_type(4)));
typedef int          i32x8 __attribute__((ext_vector_type(8)));
typedef int          i32x4 __attribute__((ext_vector_type(4)));

// logaddexp(a,b) = max + log(1 + exp(-|a-b|)); 1 v_exp_f32 + 1 v_log_f32.
__device__ __forceinline__ float lae(float a, float b) {
    float m = fmaxf(a, b);
    float d = fabsf(a - b);
    return m + __logf(1.0f + __expf(-d));
}

// Issue one TDM load of a [T_TILE rows x BLK floats] tile (row stride Dn
// elements in global memory, rows packed contiguously in LDS).
__device__ __forceinline__ void tdm_load_tile(const float* gptr, uint32_t lds_off) {
    uint64_t ga = (uint64_t)(uintptr_t)gptr;
    u32x4 g0;
    g0[0] = 1u;                                                  // count=1, user mode
    g0[1] = lds_off;                                             // lds_addr (bytes)
    g0[2] = (uint32_t)ga;                                        // global_addr[31:0]
    g0[3] = ((uint32_t)(ga >> 32) & 0x01FFFFFFu) | (2u << 30);   // addr[56:32] | type=2
    i32x8 g1;
    g1[0] = (int)(2u << 16);                                     // data_size=2 (4 bytes)
    g1[1] = (int)((uint32_t)BLK << 16);                          // tensor_dim0 lo16
    g1[2] = (int)((uint32_t)T_TILE << 16);                       // dim0 hi16=0 | tensor_dim1 lo16
    g1[3] = (int)((uint32_t)BLK << 16);                          // dim1 hi16=0 | tile_dim0
    g1[4] = (int)(uint32_t)T_TILE;                               // tile_dim1 | tile_dim2=0
    g1[5] = (int)(uint32_t)Dn;                                   // tensor_dim0_stride lo32
    g1[6] = 0;                                                   // stride0 hi | stride1 lo
    g1[7] = 0;
    i32x4 z4 = {0, 0, 0, 0};                                     // groups 2/3 unused (2D)
    i32x8 z8 = {0, 0, 0, 0, 0, 0, 0, 0};                         // trailing group (unused)
    // amdgpu-toolchain (clang-23) 6-arg form.
    __builtin_amdgcn_tensor_load_to_lds(g0, g1, z4, z4, z8, 0);
}

__global__ __launch_bounds__(BLK) void wkv_scan_kernel(
    const float* __restrict__ w, const float* __restrict__ u,
    const float* __restrict__ k, const float* __restrict__ v,
    const float* __restrict__ st0,
    float* __restrict__ wkv_out, float* __restrict__ st_out)
{
    __shared__ float lds_k[2][T_TILE * BLK];
    __shared__ float lds_v[2][T_TILE * BLK];

    const int lane = threadIdx.x;
    const int blk  = blockIdx.x;                 // 0 .. B*(D/BLK)-1
    const int b    = blk / (Dn / BLK);
    const int c0   = (blk % (Dn / BLK)) * BLK;
    const int d    = c0 + lane;

    const float wv = w[d];
    const float uv = u[d];
    float lap = st0[(b * 3 + 0) * Dn + d];
    float lam = st0[(b * 3 + 1) * Dn + d];
    float lb  = st0[(b * 3 + 2) * Dn + d];

    // state_out layout: (B, 3, T+1, D); slot t=0 carries the initial state.
    float* so_p = st_out + (size_t)(b * 3 + 0) * (Tn + 1) * Dn + d;
    float* so_m = st_out + (size_t)(b * 3 + 1) * (Tn + 1) * Dn + d;
    float* so_b = st_out + (size_t)(b * 3 + 2) * (Tn + 1) * Dn + d;
    __builtin_nontemporal_store(lap, so_p);
    __builtin_nontemporal_store(lam, so_m);
    __builtin_nontemporal_store(lb,  so_b);

    const float* kbase = k + (size_t)b * Tn * Dn + c0;
    const float* vbase = v + (size_t)b * Tn * Dn + c0;
    float*       op    = wkv_out + (size_t)b * Tn * Dn + d;   // walks t
    so_p += Dn;   // now point at t=1 slot; walk by Dn per step
    so_m += Dn;
    so_b += Dn;

    const uint32_t lk0 = (uint32_t)(uintptr_t)&lds_k[0][0];
    const uint32_t lk1 = (uint32_t)(uintptr_t)&lds_k[1][0];
    const uint32_t lv0 = (uint32_t)(uintptr_t)&lds_v[0][0];
    const uint32_t lv1 = (uint32_t)(uintptr_t)&lds_v[1][0];

    // Prime buffer 0.
    tdm_load_tile(kbase, lk0);
    tdm_load_tile(vbase, lv0);

    int buf = 0;
    for (int tile = 0; tile < NTILES; ++tile) {
        if (tile + 1 < NTILES) {
            const float* kn = kbase + (size_t)(tile + 1) * T_TILE * Dn;
            const float* vn = vbase + (size_t)(tile + 1) * T_TILE * Dn;
            tdm_load_tile(kn, buf ? lk0 : lk1);
            tdm_load_tile(vn, buf ? lv0 : lv1);
            // 2 newest in flight allowed; 2 oldest (current tile) must be done.
            __builtin_amdgcn_s_wait_tensorcnt(2);
        } else {
            __builtin_amdgcn_s_wait_tensorcnt(0);
        }
        asm volatile("" ::: "memory");   // fence LDS reads after the TDM wait

        const float* __restrict__ lk = &lds_k[buf][lane];
        const float* __restrict__ lv = &lds_v[buf][lane];

        #pragma unroll 4
        for (int tt = 0; tt < T_TILE; ++tt) {
            float kt = lk[tt * BLK];
            float vt = lv[tt * BLK];

            float vp  = fmaxf(vt, 0.0f) + WKV_EPS;
            float vm  = fmaxf(-vt, 0.0f) + WKV_EPS;
            float lvp = __logf(vp);
            float lvm = __logf(vm);

            float uk  = uv + kt;
            float den = lae(uk, lb);                 // shared denominator
            float np  = lae(uk + lvp, lap);
            float nm  = lae(uk + lvm, lam);
            float wkv = __expf(np - den) - __expf(nm - den);

            lap = lae(lap - wv, kt + lvp);
            lam = lae(lam - wv, kt + lvm);
            lb  = lae(lb  - wv, kt);

            __builtin_nontemporal_store(wkv, op);
            __builtin_nontemporal_store(lap, so_p);
            __builtin_nontemporal_store(lam, so_m);
            __builtin_nontemporal_store(lb,  so_b);
            op   += Dn;
            so_p += Dn;
            so_m += Dn;
            so_b += Dn;
        }

        buf ^= 1;
        asm volatile("" ::: "memory");   // keep next TDM issue after this tile's reads
    }
}

extern "C" void kernel_launch(void* const* d_in, const int* in_sizes, int n_in,
                              void* d_out, int out_size, void* d_ws, size_t ws_size,
                              hipStream_t stream) {
    const float* w  = (const float*)d_in[0];   // (D)
    const float* u  = (const float*)d_in[1];   // (D)
    const float* k  = (const float*)d_in[2];   // (B,T,D)
    const float* v  = (const float*)d_in[3];   // (B,T,D)
    const float* st = (const float*)d_in[4];   // (B,3,1,D)

    float* out      = (float*)d_out;
    float* wkv_out  = out;                                   // (B,T,D)
    float* st_out   = out + (size_t)Bn * Tn * Dn;            // (B,3,T+1,D)

    dim3 grid(Bn * (Dn / BLK));   // 256 single-wave workgroups
    dim3 block(BLK);              // one wave32, one channel per lane
    wkv_scan_kernel<<<grid, block, 0, stream>>>(w, u, k, v, st, wkv_out, st_out);
}